// GraphVToS_9388798509586
// MI455X (gfx1250) — compile-verified
//
#include <hip/hip_runtime.h>

typedef float v2f __attribute__((ext_vector_type(2)));
typedef float v8f __attribute__((ext_vector_type(8)));

#define BB 8
#define NN 128
#define CC 3
#define FF 64
#define KK 64
#define MM (BB * NN * CC)      // 3072 rows in the projection GEMM
#define MTILES (MM / 16)       // 192
#define NTILES (KK / 16)       // 4
#define TOTAL_TILES (MTILES * NTILES * 2)  // 1536 (x2: p_i and p_j)

// ---------------------------------------------------------------------------
// Pass A: P = VF @ W via V_WMMA_F32_16X16X4_F32 (full fp32 precision).
//   VF : (M=3072, F=64) row-major  (== vector_features flattened over b,n,c)
//   W  : w_vs (2F=128, K=64) row-major; rows [0,64) = w_i, rows [64,128) = w_j
//   pi, pj : (M, K) row-major in workspace
// One wave per 16x16 output tile, 8 waves (256 threads) per block.
// ---------------------------------------------------------------------------
__global__ __launch_bounds__(256)
void vts_proj_wmma(const float* __restrict__ vf,
                   const float* __restrict__ wvs,
                   float* __restrict__ pi,
                   float* __restrict__ pj)
{
    const int wave = threadIdx.x >> 5;
    const int tile = blockIdx.x * 8 + wave;      // 0 .. TOTAL_TILES-1
    const int which = tile & 1;                  // 0 -> p_i, 1 -> p_j
    const int tn    = (tile >> 1) & (NTILES - 1);
    const int tm    = tile >> 3;                 // 0 .. MTILES-1

    const int ln   = threadIdx.x & 31;
    const int half = ln >> 4;                    // 0 or 1
    const int r    = ln & 15;

    const float* w = wvs + which * FF * KK;      // w_i or w_j block of w_vs
    const int m = tm * 16 + r;                   // A-matrix row for this lane
    const int n = tn * 16 + r;                   // B-matrix col for this lane

    v8f acc = {};
    #pragma unroll
    for (int kk = 0; kk < FF; kk += 4) {
        const int f0 = kk + 2 * half;            // K index held by VGPR 0
        // A 16x4 layout: lane r(+16*half) holds K = f0 (vgpr0), f0+1 (vgpr1)
        v2f a = *(const v2f*)(vf + m * FF + f0);
        // B 4x16 layout mirrored: lane holds rows f0, f0+1 at column n
        v2f b;
        b.x = w[(f0 + 0) * KK + n];
        b.y = w[(f0 + 1) * KK + n];
        acc = __builtin_amdgcn_wmma_f32_16x16x4_f32(
            /*neg_a=*/false, a, /*neg_b=*/false, b,
            /*c_mod=*/(short)0, acc, /*reuse_a=*/false, /*reuse_b=*/false);
    }

    float* p = which ? pj : pi;
    #pragma unroll
    for (int v = 0; v < 8; ++v) {
        const int mo = tm * 16 + v + 8 * half;   // C/D layout: M = v + 8*half
        p[mo * KK + tn * 16 + r] = acc[v];
    }
}

// ---------------------------------------------------------------------------
// Pass B: streaming fused epilogue (HBM-store-bound part).
//   s[b,i,j,k] = relu( sum_c d[b,i,j,c]*(pi[b,i,c,k] + pj[b,j,c,k])
//                      + (sum_c d[b,i,j,c]) * b_vs[k] )
// One wave per (b,i,j) row; lane writes k = 2*lane, 2*lane+1 as one float2
// -> each 256B output row is a single fully-coalesced wave store.
// ---------------------------------------------------------------------------
__global__ __launch_bounds__(256)
void vts_epilogue(const float* __restrict__ dist,  // (B,N,N,C)
                  const float* __restrict__ pi,    // (B,N,C,K)
                  const float* __restrict__ pj,    // (B,N,C,K)
                  const float* __restrict__ bvs,   // (K)
                  float* __restrict__ out)         // (B,N,N,K)
{
    const int lane = threadIdx.x & 31;
    const int wave = threadIdx.x >> 5;
    const int row  = blockIdx.x * 8 + wave;        // flat (b,i,j), 131072 rows
    const int j = row & (NN - 1);
    const int i = (row >> 7) & (NN - 1);
    const int b = row >> 14;

    const float* dptr = dist + (size_t)row * CC;
    const float d0 = dptr[0];
    const float d1 = dptr[1];
    const float d2 = dptr[2];
    const float ds = d0 + d1 + d2;

    const float* piB = pi + ((b * NN + i) * CC) * KK;
    const float* pjB = pj + ((b * NN + j) * CC) * KK;

    const int k = lane * 2;
    const v2f pi0 = *(const v2f*)(piB + 0 * KK + k);
    const v2f pi1 = *(const v2f*)(piB + 1 * KK + k);
    const v2f pi2 = *(const v2f*)(piB + 2 * KK + k);
    const v2f pj0 = *(const v2f*)(pjB + 0 * KK + k);
    const v2f pj1 = *(const v2f*)(pjB + 1 * KK + k);
    const v2f pj2 = *(const v2f*)(pjB + 2 * KK + k);
    const v2f bv  = *(const v2f*)(bvs + k);

    v2f s;
    s.x = fmaf(d0, pi0.x + pj0.x, fmaf(d1, pi1.x + pj1.x,
          fmaf(d2, pi2.x + pj2.x, ds * bv.x)));
    s.y = fmaf(d0, pi0.y + pj0.y, fmaf(d1, pi1.y + pj1.y,
          fmaf(d2, pi2.y + pj2.y, ds * bv.y)));
    s.x = fmaxf(s.x, 0.0f);
    s.y = fmaxf(s.y, 0.0f);

    *(v2f*)(out + (size_t)row * KK + k) = s;
}

// ---------------------------------------------------------------------------
extern "C" void kernel_launch(void* const* d_in, const int* in_sizes, int n_in,
                              void* d_out, int out_size, void* d_ws, size_t ws_size,
                              hipStream_t stream) {
    const float* vf   = (const float*)d_in[0];   // vector_features (B,N,C,F)
    const float* dist = (const float*)d_in[1];   // distances       (B,N,N,C)
    const float* wvs  = (const float*)d_in[2];   // w_vs            (2F,K)
    const float* bvs  = (const float*)d_in[3];   // b_vs            (K)
    float* out = (float*)d_out;

    float* pi = (float*)d_ws;                    // (M,K) = 786 KB
    float* pj = pi + (size_t)MM * KK;            // (M,K)

    // Pass A: projection GEMMs via f32 WMMA (1536 tiles, 8 waves/block)
    vts_proj_wmma<<<TOTAL_TILES / 8, 256, 0, stream>>>(vf, wvs, pi, pj);

    // Pass B: fused bandwidth-bound epilogue (131072 rows, 8 rows/block)
    vts_epilogue<<<(BB * NN * NN) / 8, 256, 0, stream>>>(dist, pi, pj, bvs, out);
}